// QNN_56676388438509
// MI455X (gfx1250) — compile-verified
//
#include <hip/hip_runtime.h>

typedef __attribute__((ext_vector_type(2))) float v2f;
typedef __attribute__((ext_vector_type(8))) float v8f;

// Extended weight matrix W_ext (2 x 10): columns 0..8 = W, column 9 = bias
// (bias folded into the GEMM via a constant-1 feature). Zero-padded in N and K.
__device__ __forceinline__ float wext(const float* __restrict__ W,
                                      const float* __restrict__ b,
                                      int n, int k) {
    if (n >= 2) return 0.0f;
    if (k < 9)  return W[n * 9 + k];
    if (k == 9) return b[n];
    return 0.0f;  // K padding (k = 10, 11)
}

__global__ __launch_bounds__(256) void qnn_wmma_kernel(
    const float* __restrict__ x,        // (B, 8)
    const float* __restrict__ enc_off,  // (8)
    const float* __restrict__ enc_sc,   // (8)
    const float* __restrict__ W,        // (2, 9)
    const float* __restrict__ bias,     // (2)
    float* __restrict__ out,            // (B, 2)
    int B)
{
    const int lane = threadIdx.x & 31;
    const int wave = blockIdx.x * (blockDim.x >> 5) + (threadIdx.x >> 5);
    const int base = wave * 16;          // 16 batch rows per wave
    if (base >= B) return;               // wave-uniform: EXEC stays all-ones

    // GF(2) dependency masks of the measured bits on the encoding bits:
    // push 8 bit-variables through 3 layers of the CNOT ring. Fully
    // compile-time foldable (constant trip counts).
    unsigned m[8];
#pragma unroll
    for (int j = 0; j < 8; ++j) m[j] = 1u << j;
#pragma unroll
    for (int l = 0; l < 3; ++l)
#pragma unroll
        for (int i = 0; i < 8; ++i) m[(i + 1) & 7] ^= m[i];
    unsigned pm = 0;                      // parity observable Z0..Z7
#pragma unroll
    for (int j = 0; j < 8; ++j) pm ^= m[j];

    const int  r   = lane & 15;           // row within the 16-row tile
    const bool hi  = lane >= 16;          // upper lane half carries K=2,3 slices
    const long long row = base + r;       // both halves redundantly compute same row

    const float4* xp = (const float4*)(x + row * 8);
    const float4 x0 = xp[0];
    const float4 x1 = xp[1];
    const float xv[8] = {x0.x, x0.y, x0.z, x0.w, x1.x, x1.y, x1.z, x1.w};

    // RZ phases cancel in |psi|^2 -> <Z_i> = prod_{j in S_i} cos(a_j)
    float cj[8];
#pragma unroll
    for (int j = 0; j < 8; ++j)
        cj[j] = cosf(enc_off[j] + enc_sc[j] * xv[j]);

    float f[9];
#pragma unroll
    for (int i = 0; i < 9; ++i) {
        const unsigned msk = (i < 8) ? m[i] : pm;
        float p = 1.0f;
#pragma unroll
        for (int j = 0; j < 8; ++j)
            p *= ((msk >> j) & 1u) ? cj[j] : 1.0f;
        f[i] = p;
    }

    // A fragments: 16x4 f32 tiles of the (16 x 10->12) feature matrix.
    // Lanes 0-15 hold K = 4c+0 (v.x), 4c+1 (v.y); lanes 16-31 hold K = 4c+2, 4c+3.
    v2f A0, A1, A2;
    A0.x = hi ? f[2] : f[0];   A0.y = hi ? f[3] : f[1];
    A1.x = hi ? f[6] : f[4];   A1.y = hi ? f[7] : f[5];
    A2.x = hi ? 0.0f : f[8];   A2.y = hi ? 0.0f : 1.0f;   // k=9 feature == 1 (bias)

    // B fragments: 4x16 f32 tiles of W_ext^T (10 x 16, zero-padded), same K split.
    const int n  = lane & 15;             // output column (only 0,1 are live)
    const int kb = hi ? 2 : 0;
    v2f B0, B1, B2;
    B0.x = wext(W, bias, n, 0 + kb);  B0.y = wext(W, bias, n, 1 + kb);
    B1.x = wext(W, bias, n, 4 + kb);  B1.y = wext(W, bias, n, 5 + kb);
    B2.x = wext(W, bias, n, 8 + kb);  B2.y = wext(W, bias, n, 9 + kb);

    // D = A x B + C, chained over the three K=4 chunks (f32 path: exact).
    v8f acc = {};
    acc = __builtin_amdgcn_wmma_f32_16x16x4_f32(false, A0, false, B0,
                                                (short)0, acc, false, false);
    acc = __builtin_amdgcn_wmma_f32_16x16x4_f32(false, A1, false, B1,
                                                (short)0, acc, false, false);
    acc = __builtin_amdgcn_wmma_f32_16x16x4_f32(false, A2, false, B2,
                                                (short)0, acc, false, false);

    // D layout: VGPR rr = row M=rr (lanes 0-15, N=lane) / M=rr+8 (lanes 16-31).
    // Lanes 0,1,16,17 hold the two live output columns for all 16 rows.
    if (n < 2) {
        const int moff = hi ? 8 : 0;
#pragma unroll
        for (int rr = 0; rr < 8; ++rr)
            out[(long long)(base + moff + rr) * 2 + n] = acc[rr];
    }
}

extern "C" void kernel_launch(void* const* d_in, const int* in_sizes, int n_in,
                              void* d_out, int out_size, void* d_ws, size_t ws_size,
                              hipStream_t stream) {
    const float* x   = (const float*)d_in[0];
    // d_in[1] (weights) is provably unused: RZ phases are unit-modulus and the
    // CNOT ring is a basis permutation, so |amplitudes|^2 are weight-independent.
    const float* off = (const float*)d_in[2];
    const float* sc  = (const float*)d_in[3];
    const float* W   = (const float*)d_in[4];
    const float* b   = (const float*)d_in[5];
    float* out = (float*)d_out;

    const int B      = in_sizes[0] / 8;        // 65536
    const int waves  = (B + 15) / 16;          // 16 rows per wave
    const int blocks = (waves + 7) / 8;        // 8 waves (256 thr) per block

    qnn_wmma_kernel<<<blocks, 256, 0, stream>>>(x, off, sc, W, b, out, B);
}